// QuantumLayer_15281493639355
// MI455X (gfx1250) — compile-verified
//
#include <hip/hip_runtime.h>

// MI455X / gfx1250, wave32. One workgroup = one batch sample.
// State vector lives entirely in LDS (128 KB << 320 KB/WGP, 2 blocks/WGP).
// Single-qubit gates fused pairwise (4 amps in registers per pass) to halve
// LDS sweeps + barriers. Final <Z_q> reduction is a 16x16xK f32 GEMM on the
// matrix core (V_WMMA_F32_16X16X4_F32) with branchless in-register signs.

typedef __attribute__((ext_vector_type(2))) float v2f;
typedef __attribute__((ext_vector_type(8))) float v8f;

#define NQ   14
#define NST  (1 << NQ)      // 16384 amplitudes
#define NL   3
#define TPB  256            // 8 wave32s

// RY pair update: [[c,-s],[s,c]] (real coefficients, applied to re & im)
__device__ __forceinline__ void ry2(float2& a0, float2& a1, float c, float s) {
    const float2 n0 = make_float2(c * a0.x - s * a1.x, c * a0.y - s * a1.y);
    const float2 n1 = make_float2(s * a0.x + c * a1.x, s * a0.y + c * a1.y);
    a0 = n0; a1 = n1;
}

// RX pair update: [[c,-is],[-is,c]]
__device__ __forceinline__ void rx2(float2& a0, float2& a1, float c, float s) {
    const float2 n0 = make_float2(c * a0.x + s * a1.y, c * a0.y - s * a1.x);
    const float2 n1 = make_float2(c * a1.x + s * a0.y, c * a1.y - s * a0.x);
    a0 = n0; a1 = n1;
}

__global__ __launch_bounds__(TPB) void qsim_wmma_kernel(
    const float* __restrict__ x,       // (B, 14)
    const float* __restrict__ w,       // (3, 14)
    float* __restrict__ out)           // (B, 14)
{
    __shared__ float2 st[NST];         // 128 KB amplitude state (re, im)
    __shared__ float  red[8][16][16];  // 8 KB per-wave WMMA partials

    const int b = blockIdx.x;
    const int t = threadIdx.x;

    // ---- init |0...0> ----
    for (int i = t; i < NST; i += TPB) st[i] = make_float2(0.f, 0.f);
    if (t == 0) st[0] = make_float2(1.f, 0.f);
    __syncthreads();

    // ---- AngleEmbedding: RY(tanh(x)*pi), fused in qubit pairs (qa, qa+1) ----
    for (int qa = 0; qa < NQ; qa += 2) {
        const int qb = qa + 1;
        const float tha = tanhf(x[b * NQ + qa]) * 3.14159265358979323846f * 0.5f;
        const float thb = tanhf(x[b * NQ + qb]) * 3.14159265358979323846f * 0.5f;
        const float ca = cosf(tha), sa = sinf(tha);
        const float cb = cosf(thb), sb = sinf(thb);
        const int pa = NQ - 1 - qa;          // higher bit position
        const int pb = NQ - 1 - qb;          // lower bit position
        const int lowb = (1 << pb) - 1, lowa = (1 << pa) - 1;
        const int iA = 1 << pa, iB = 1 << pb;
        for (int g = t; g < (NST >> 2); g += TPB) {
            const int t1   = ((g  & ~lowb) << 1) | (g  & lowb);  // insert 0 at pb
            const int base = ((t1 & ~lowa) << 1) | (t1 & lowa);  // insert 0 at pa
            float2 a00 = st[base],      a01 = st[base | iB];
            float2 a10 = st[base | iA], a11 = st[base | iA | iB];
            ry2(a00, a10, ca, sa); ry2(a01, a11, ca, sa);        // gate on qa
            ry2(a00, a01, cb, sb); ry2(a10, a11, cb, sb);        // gate on qb
            st[base]      = a00; st[base | iB]      = a01;
            st[base | iA] = a10; st[base | iA | iB] = a11;
        }
        __syncthreads();
    }

    // ---- BasicEntanglerLayers ----
    for (int l = 0; l < NL; ++l) {
        // RX(w[l,q]) on every wire, fused in pairs
        for (int qa = 0; qa < NQ; qa += 2) {
            const int qb = qa + 1;
            const float tha = w[l * NQ + qa] * 0.5f;
            const float thb = w[l * NQ + qb] * 0.5f;
            const float ca = cosf(tha), sa = sinf(tha);
            const float cb = cosf(thb), sb = sinf(thb);
            const int pa = NQ - 1 - qa, pb = NQ - 1 - qb;
            const int lowb = (1 << pb) - 1, lowa = (1 << pa) - 1;
            const int iA = 1 << pa, iB = 1 << pb;
            for (int g = t; g < (NST >> 2); g += TPB) {
                const int t1   = ((g  & ~lowb) << 1) | (g  & lowb);
                const int base = ((t1 & ~lowa) << 1) | (t1 & lowa);
                float2 a00 = st[base],      a01 = st[base | iB];
                float2 a10 = st[base | iA], a11 = st[base | iA | iB];
                rx2(a00, a10, ca, sa); rx2(a01, a11, ca, sa);
                rx2(a00, a01, cb, sb); rx2(a10, a11, cb, sb);
                st[base]      = a00; st[base | iB]      = a01;
                st[base | iA] = a10; st[base | iA | iB] = a11;
            }
            __syncthreads();
        }
        // CNOT ring: control q -> target (q+1)%14. Pure swaps on control=1 block.
        for (int q = 0; q < NQ; ++q) {
            const int pc = NQ - 1 - q;
            const int pt = NQ - 1 - ((q + 1) % NQ);
            const int pl = pc < pt ? pc : pt;
            const int ph = pc < pt ? pt : pc;
            const int lowl = (1 << pl) - 1;
            const int lowh = (1 << ph) - 1;
            for (int idx = t; idx < (NST >> 2); idx += TPB) {
                int t1 = ((idx & ~lowl) << 1) | (idx & lowl);   // insert 0 at pl
                int i  = ((t1  & ~lowh) << 1) | (t1  & lowh);   // insert 0 at ph
                i |= (1 << pc);                                 // control = 1, target = 0
                const int j = i | (1 << pt);
                const float2 tmp = st[i];
                st[i] = st[j];
                st[j] = tmp;
            }
            __syncthreads();
        }
    }

    // ---- <Z_q> via V_WMMA_F32_16X16X4_F32 ----
    // Probs viewed as M=16 chunks x K=1024.  State index s = m*1024 + k:
    //   chunk bits m = qubits 0..3, k bits = qubits 4..13.
    // B columns: n in 0..9  -> sign of qubit (n+4) = 1-2*((k >> (9-n)) & 1)
    //            n == 10    -> 1.0  (row sums, recover qubits 0..3)
    //            n >= 11    -> 0.0
    // Branchless: shift=31 makes the extracted bit 0 (sign=+1) for n>=10,
    // scale=0 kills columns >= 11.  Both are per-lane loop invariants.
    const int lane  = t & 31;
    const int wave  = t >> 5;
    const int mrow  = lane & 15;            // A: M = lane % 16
    const int khalf = (lane >> 4) << 1;     // A/B: lanes 16..31 carry K = 2,3
    const int ncol  = lane & 15;            // B/D: N = lane % 16
    const int   sh    = (ncol < 10) ? (9 - ncol) : 31;
    const float scale = (ncol <= 10) ? 1.f : 0.f;

    v8f acc = {};
    for (int kb = wave * 4; kb < 1024; kb += 8 * 4) {   // 32 WMMAs per wave
        const int k0 = kb + khalf;
        const float2 c0 = st[mrow * 1024 + k0];
        const float2 c1 = st[mrow * 1024 + k0 + 1];
        v2f av, bv;
        av[0] = c0.x * c0.x + c0.y * c0.y;              // |amp|^2 on the fly
        av[1] = c1.x * c1.x + c1.y * c1.y;
        bv[0] = scale * (1.f - 2.f * (float)((k0 >> sh) & 1));
        bv[1] = scale * (1.f - 2.f * (float)(((k0 + 1) >> sh) & 1));
        // 8 args: (neg_a, A, neg_b, B, c_mod, C, reuse_a, reuse_b)
        acc = __builtin_amdgcn_wmma_f32_16x16x4_f32(
            false, av, false, bv, (short)0, acc, false, false);
    }

    // dump per-wave partial D (layout: VGPR v, lane -> M = v + 8*(lane>=16), N = lane%16)
    {
        const int mbase = (lane >> 4) << 3;
        #pragma unroll
        for (int v = 0; v < 8; ++v)
            red[wave][mbase + v][ncol] = acc[v];
    }
    __syncthreads();

    // reduce across the 8 waves: 256 threads cover the 16x16 tile
    {
        const int m = t >> 4, n = t & 15;
        float sum = 0.f;
        #pragma unroll
        for (int wv = 0; wv < 8; ++wv) sum += red[wv][m][n];
        red[0][m][n] = sum;   // only this thread reads red[0][m][n] (wv=0) before writing
    }
    __syncthreads();

    // finalize the 14 expectation values
    if (t < NQ) {
        float e = 0.f;
        if (t >= 4) {
            const int n = t - 4;
            #pragma unroll
            for (int m = 0; m < 16; ++m) e += red[0][m][n];
        } else {
            // qubits 0..3: sign lives in the chunk index; use row sums (col 10)
            #pragma unroll
            for (int m = 0; m < 16; ++m) {
                const float sg = 1.f - 2.f * (float)((m >> (3 - t)) & 1);
                e += sg * red[0][m][10];
            }
        }
        out[b * NQ + t] = e;
    }
}

extern "C" void kernel_launch(void* const* d_in, const int* in_sizes, int n_in,
                              void* d_out, int out_size, void* d_ws, size_t ws_size,
                              hipStream_t stream) {
    const float* x = (const float*)d_in[0];   // (B, 14) float32
    const float* w = (const float*)d_in[1];   // (3, 14) float32
    float* out = (float*)d_out;               // (B, 14) float32
    const int batch = in_sizes[0] / NQ;       // 256
    qsim_wmma_kernel<<<batch, TPB, 0, stream>>>(x, w, out);
}